// StructurallyGuidedDeformConv1d_87256555586170
// MI455X (gfx1250) — compile-verified
//
#include <hip/hip_runtime.h>
#include <hip/hip_bf16.h>

// ---------------------------------------------------------------------------
// Deformable 1D conv on gfx1250:
//   sampled[b,c,k,l] = lerp(x[b,c], p=tap(k)+l+offset[b,k,l])   (zeros pad)
//   out[b,o,l]       = sum_{c,k} sampled[b,c,k,l] * W[o,c,k] + bias[o]
// GEMM per batch: (C_OUT x RED) x (RED x L_OUT), RED = C_IN*K = 1280,
// B-operand (sampled) generated on the fly into LDS (L2-resident gathers),
// MACs via V_WMMA_F32_16X16X4_F32 (fp32-exact vs. the fp32 reference).
// 8 waves = 4(M) x 2(N) wave grid, each wave owns a 32x32 tile
// (2x2 fragments). Both LDS tiles are stored K-contiguous ([m][r] / [l][r])
// so every fragment load is one ds_load_2addr of two consecutive floats
// straight into an adjacent VGPR pair (no repacking movs).
// ---------------------------------------------------------------------------

#define BATCH 8
#define CIN   256
#define COUT  256
#define KTAPS 5
#define LIN   4096
#define LOUT  4096
#define RED   (CIN * KTAPS)   // 1280

#define M_TILE 128            // C_out per block
#define N_TILE 64             // L_out per block
#define KC     32             // reduction chunk
#define APAD   1
#define BPAD   1

typedef __attribute__((ext_vector_type(2))) float v2f;
typedef __attribute__((ext_vector_type(8))) float v8f;

__global__ __launch_bounds__(256)
void sgdc1d_wmma_kernel(const float* __restrict__ x,
                        const float* __restrict__ offset,
                        const float* __restrict__ weight,
                        const float* __restrict__ bias,
                        float* __restrict__ out)
{
    __shared__ float A_lds[M_TILE][KC + APAD];      // weight tile  [m][r]
    __shared__ float B_lds[N_TILE][KC + BPAD];      // sampled tile [l][r]
    __shared__ int   s_i0[KTAPS][N_TILE];
    __shared__ int   s_i1[KTAPS][N_TILE];
    __shared__ float s_w0[KTAPS][N_TILE];
    __shared__ float s_w1[KTAPS][N_TILE];

    const int tid     = threadIdx.x;
    const int tiles_n = LOUT / N_TILE;              // 64
    const int tiles_m = COUT / M_TILE;              // 2
    const int bidx    = blockIdx.x;
    const int b       = bidx / (tiles_n * tiles_m);
    const int rem     = bidx % (tiles_n * tiles_m);
    const int m0      = (rem / tiles_n) * M_TILE;
    const int l0      = (rem % tiles_n) * N_TILE;

    // ---- per-(k,l) sampling parameters (channel independent) ----
    for (int kl = tid; kl < KTAPS * N_TILE; kl += 256) {
        int   k  = kl / N_TILE;
        int   l  = kl - k * N_TILE;
        float p  = (float)(k - KTAPS / 2) + (float)(l0 + l)
                 + offset[((size_t)b * KTAPS + k) * LOUT + l0 + l];
        float fl = floorf(p);
        int   i0 = (int)fl;
        int   i1 = i0 + 1;
        float w1 = p - fl;
        float v0 = (i0 >= 0 && i0 < LIN) ? 1.0f : 0.0f;
        float v1 = (i1 >= 0 && i1 < LIN) ? 1.0f : 0.0f;
        s_i0[k][l] = min(max(i0, 0), LIN - 1);
        s_i1[k][l] = min(max(i1, 0), LIN - 1);
        s_w0[k][l] = (1.0f - w1) * v0;
        s_w1[k][l] = w1 * v1;
    }
    __syncthreads();

    const int wave   = tid >> 5;
    const int lane   = tid & 31;
    const int m_w    = wave >> 1;        // 0..3 -> 32-row M strip
    const int n_w    = wave & 1;         // 0..1 -> 32-col N strip
    const int mBase  = m_w * 32;         // wave M origin (2 frags: +0, +16)
    const int nBase  = n_w * 32;         // wave N origin (2 frags: +0, +16)
    const int laneHi = lane >> 4;        // 0/1: K pair select & M+8 select
    const int laneLo = lane & 15;

    v8f acc00 = {};  // (M+0 , N+0 )
    v8f acc01 = {};  // (M+0 , N+16)
    v8f acc10 = {};  // (M+16, N+0 )
    v8f acc11 = {};  // (M+16, N+16)

    const float* __restrict__ xb = x + (size_t)b * CIN * LIN;

    for (int r0 = 0; r0 < RED; r0 += KC) {
        __syncthreads();   // previous chunk's WMMA reads done before refill

        // ---- stage A: A_lds[m][r] = weight[(m0+m)*RED + r0 + r] ----
        {
            int r  = tid & (KC - 1);      // 0..31 consecutive -> coalesced
            int mb = tid >> 5;            // 0..7
            #pragma unroll
            for (int i = 0; i < M_TILE / 8; ++i) {
                int m = mb + i * 8;
                A_lds[m][r] = weight[(size_t)(m0 + m) * RED + (r0 + r)];
            }
        }

        // ---- stage B: fused gather+lerp, B_lds[l][r] = sampled[r0+r][l0+l]
        {
            int l  = tid & (N_TILE - 1);  // 0..63
            int rb = tid >> 6;            // 0..3
            #pragma unroll
            for (int i = 0; i < KC / 4; ++i) {
                int r  = rb + i * 4;
                int rr = r0 + r;
                int c  = rr / KTAPS;
                int k  = rr - c * KTAPS;
                float v = xb[(size_t)c * LIN + s_i0[k][l]] * s_w0[k][l]
                        + xb[(size_t)c * LIN + s_i1[k][l]] * s_w1[k][l];
                B_lds[l][r] = v;
            }
        }
        __syncthreads();

        // ---- 8 WMMA steps of K=4; 2x2 fragments -> 4 WMMAs per step ----
        #pragma unroll
        for (int kk = 0; kk < KC; kk += 4) {
            int ka = kk + 2 * laneHi;     // lanes 0-15: K={kk,kk+1}; 16-31: {+2,+3}
            v2f a0, a1;
            a0.x = A_lds[mBase + laneLo][ka];
            a0.y = A_lds[mBase + laneLo][ka + 1];
            a1.x = A_lds[mBase + 16 + laneLo][ka];
            a1.y = A_lds[mBase + 16 + laneLo][ka + 1];
            v2f bf0, bf1;
            bf0.x = B_lds[nBase + laneLo][ka];
            bf0.y = B_lds[nBase + laneLo][ka + 1];
            bf1.x = B_lds[nBase + 16 + laneLo][ka];
            bf1.y = B_lds[nBase + 16 + laneLo][ka + 1];
            acc00 = __builtin_amdgcn_wmma_f32_16x16x4_f32(
                        false, a0, false, bf0, (short)0, acc00, false, false);
            acc01 = __builtin_amdgcn_wmma_f32_16x16x4_f32(
                        false, a0, false, bf1, (short)0, acc01, false, false);
            acc10 = __builtin_amdgcn_wmma_f32_16x16x4_f32(
                        false, a1, false, bf0, (short)0, acc10, false, false);
            acc11 = __builtin_amdgcn_wmma_f32_16x16x4_f32(
                        false, a1, false, bf1, (short)0, acc11, false, false);
        }
    }

    // ---- epilogue: D layout VGPR v -> M = v + 8*laneHi, N = laneLo ----
    #pragma unroll
    for (int v = 0; v < 8; ++v) {
        int    mA   = m0 + mBase + v + 8 * laneHi;       // fragments (M+0, *)
        int    mB   = mA + 16;                           // fragments (M+16, *)
        float  bvA  = bias[mA];
        float  bvB  = bias[mB];
        size_t rowA = ((size_t)b * COUT + mA) * LOUT + l0 + nBase;
        size_t rowB = ((size_t)b * COUT + mB) * LOUT + l0 + nBase;
        out[rowA + laneLo]      = acc00[v] + bvA;
        out[rowA + 16 + laneLo] = acc01[v] + bvA;
        out[rowB + laneLo]      = acc10[v] + bvB;
        out[rowB + 16 + laneLo] = acc11[v] + bvB;
    }
}

extern "C" void kernel_launch(void* const* d_in, const int* in_sizes, int n_in,
                              void* d_out, int out_size, void* d_ws, size_t ws_size,
                              hipStream_t stream) {
    (void)in_sizes; (void)n_in; (void)out_size; (void)d_ws; (void)ws_size;
    const float* x      = (const float*)d_in[0];
    const float* offset = (const float*)d_in[1];
    const float* weight = (const float*)d_in[2];
    const float* bias   = (const float*)d_in[3];
    float*       out    = (float*)d_out;

    const int tiles = (COUT / M_TILE) * (LOUT / N_TILE);   // 2 * 64 = 128
    dim3 grid(BATCH * tiles);                              // 1024 blocks
    dim3 block(256);
    sgdc1d_wmma_kernel<<<grid, block, 0, stream>>>(x, offset, weight, bias, out);
}